// GraphTextLM_8134668058635
// MI455X (gfx1250) — compile-verified
//
#include <hip/hip_runtime.h>
#include <hip/hip_bf16.h>
#include <math.h>

// ---------------------------------------------------------------------------
// Problem constants (from reference)
// ---------------------------------------------------------------------------
#define NNODES 2048
#define NEDGES 65536
#define INCH   256
#define HCH    256
#define NREL   5
#define NBASES 30
#define PHD    768
#define NCLS   10
#define NHEADS 8
#define DH     96          // PHD / NHEADS
#define EPSV   1e-5f

// ---------------------------------------------------------------------------
// Input index map (recursive insertion-order flattening of setup_inputs())
// ---------------------------------------------------------------------------
// 0 gnn_input_features  1 edge_index  2 edge_type  3 plm_embeds  4 text_mask
// rgcn{i}: 5+7i {bases,comp,root,bias}; gnorm{i}: 5+7i+4 {w,b,alpha}
#define RGBASE   5
#define RES1W   33
#define RES2W   35
// RES3 (37,38) dead in forward
#define MSF_SW  39
#define MSF_G   40
#define MSF_B   41
#define MSF_P0  42   // proj i: W at 42+2i, b at 43+2i
#define G2T     50   // qW qb kW kb vW vb oW ob
#define T2G     58
#define FUSW    66
#define FUSB    67
#define FLNG    68
#define FLNB    69
#define C1W     70
#define C1B     71
#define C2W     72
#define C2B     73

typedef float v2f __attribute__((ext_vector_type(2)));
typedef float v8f __attribute__((ext_vector_type(8)));
typedef unsigned int v4u __attribute__((ext_vector_type(4)));
typedef int v8i __attribute__((ext_vector_type(8)));
typedef int v4i __attribute__((ext_vector_type(4)));

#if __has_builtin(__builtin_amdgcn_tensor_load_to_lds) && __has_builtin(__builtin_amdgcn_s_wait_tensorcnt)
#define USE_TDM 1
#else
#define USE_TDM 0
#endif

__device__ __forceinline__ float gelu_f(float x) {
    return 0.5f * x * (1.0f + erff(x * 0.70710678118654752440f));
}

#if USE_TDM
__device__ __forceinline__ unsigned lds_byte_off(const void* p) {
    // generic pointer to LDS: low 32 bits are the LDS byte address
    return (unsigned)(unsigned long long)(size_t)p;
}

// 2D TDM load: tile_y rows of tile_x f32 elements, global row stride (f32 units),
// with optional LDS padding of (pad_amount_code+1... encoded) DWORDs every
// 2^(pad_interval_code+1) DWORDs, per CDNA5 D# spec.
__device__ __forceinline__ void tdm_load_2d(unsigned lds_addr, const float* gptr,
                                            unsigned tile_x, unsigned tile_y,
                                            unsigned row_stride,
                                            unsigned pad_interval_code,
                                            unsigned pad_amount_code)
{
    unsigned long long ga = (unsigned long long)(size_t)(const void*)gptr;
    v4u g0;
    g0[0] = 1u;                                             // count=1 (valid), user mode
    g0[1] = lds_addr;                                       // lds_addr (bytes)
    g0[2] = (unsigned)(ga & 0xffffffffu);                   // global_addr[31:0]
    g0[3] = (unsigned)((ga >> 32) & 0x01ffffffu) | (2u << 30);  // addr[56:32] | type=2

    const unsigned td0 = 0x7fffffffu;                       // tensor dims large: no OOB clip
    const unsigned td1 = 0x7fffffffu;
    unsigned long long s0 = (unsigned long long)row_stride; // tensor_dim0_stride (f32 units)
    v8i g1;
    g1[0] = (int)((2u << 16)                                // data_size = 4 bytes
                | (1u << 20)                                // pad_enable
                | (pad_interval_code << 22)
                | (pad_amount_code << 25));
    g1[1] = (int)((td0 & 0xffffu) << 16);                   // atomic_barrier=0 | td0[15:0]
    g1[2] = (int)(((td0 >> 16) & 0xffffu) | ((td1 & 0xffffu) << 16));
    g1[3] = (int)(((td1 >> 16) & 0xffffu) | ((tile_x & 0xffffu) << 16));  // tile_dim0
    g1[4] = (int)(tile_y & 0xffffu);                        // tile_dim1 | tile_dim2=0
    g1[5] = (int)(unsigned)(s0 & 0xffffffffu);              // stride0[31:0]
    g1[6] = (int)(unsigned)((s0 >> 32) & 0xffffu);          // stride0[47:32] | stride1=0
    g1[7] = 0;
    v4i g2 = {0, 0, 0, 0};
    v4i g3 = {0, 0, 0, 0};
#if defined(__AMDGCN__)
#if __clang_major__ >= 23
    v8i g4 = {0, 0, 0, 0, 0, 0, 0, 0};
    __builtin_amdgcn_tensor_load_to_lds(g0, g1, g2, g3, g4, 0);
#else
    __builtin_amdgcn_tensor_load_to_lds(g0, g1, g2, g3, 0);
#endif
#endif
}
#endif  // USE_TDM

// ---------------------------------------------------------------------------
// WMMA fp32 GEMM:  C = scale*(A[MxK] @ B[KxN] + bias) (+residual) (+=C) (gelu)
// Macro tile 128x64, 256 threads = 8 waves, wave owns 16 rows x 64 cols.
// A/B tiles staged into LDS by the Tensor Data Mover (double buffered),
// synchronized with s_wait_tensorcnt. Requires M%128==0, N%64==0, K%16==0.
// ---------------------------------------------------------------------------
#define BM 128
#define BN 64
#define BK 16
#define AS_LD 17   // [m][k] layout, 16 + 1 pad DWORD per row (TDM pad: 2^(3+1)=16 dw, +1 dw)
#define BS_LD 68   // [k][n] layout, 64 + 4 pad DWORDs per row (TDM pad: 2^(5+1)=64 dw, +4 dw)

__global__ __launch_bounds__(256) void gemm_wmma_f32(
    const float* __restrict__ A, int lda,
    const float* __restrict__ B, int ldb,
    float* __restrict__ C, int ldc,
    const float* __restrict__ bias,
    const float* __restrict__ residual, int ldr,
    const float* __restrict__ scale_ptr,
    int K, int flags)  // flags: bit0 accumulate into C, bit1 gelu
{
    __shared__ float As[2][BM * AS_LD];
    __shared__ float Bs[2][BK * BS_LD];

    const int tid  = threadIdx.x;
    const int lane = tid & 31;
    const int wave = tid >> 5;      // 0..7
    const int ln16 = lane & 15;
    const int half = lane >> 4;     // 0/1

    const int row0 = blockIdx.y * BM;
    const int col0 = blockIdx.x * BN;
    const int nk = K / BK;

    v8f acc[4] = {};

#if USE_TDM
    if (wave == 0) {
        tdm_load_2d(lds_byte_off(&As[0][0]), A + (size_t)row0 * lda, BK, BM, (unsigned)lda, 3u, 0u);
        tdm_load_2d(lds_byte_off(&Bs[0][0]), B + col0, BN, BK, (unsigned)ldb, 5u, 3u);
    }
#endif

    for (int kt = 0; kt < nk; kt++) {
        const int cur = kt & 1;
#if USE_TDM
        if (wave == 0) {
            if (kt + 1 < nk) {
                const int k0n = (kt + 1) * BK;
                tdm_load_2d(lds_byte_off(&As[cur ^ 1][0]),
                            A + (size_t)row0 * lda + k0n, BK, BM, (unsigned)lda, 3u, 0u);
                tdm_load_2d(lds_byte_off(&Bs[cur ^ 1][0]),
                            B + (size_t)k0n * ldb + col0, BN, BK, (unsigned)ldb, 5u, 3u);
                __builtin_amdgcn_s_wait_tensorcnt(2);   // oldest pair (this iter's data) done
            } else {
                __builtin_amdgcn_s_wait_tensorcnt(0);
            }
        }
#else
        {   // manual staging fallback
            const int k0 = kt * BK;
            const int am = tid >> 2;            // 0..63
            const int ak = (tid & 3) * 4;       // 0,4,8,12
            const float4 a0 = *(const float4*)(A + (size_t)(row0 + am) * lda + (k0 + ak));
            const float4 a1 = *(const float4*)(A + (size_t)(row0 + am + 64) * lda + (k0 + ak));
            As[cur][am * AS_LD + ak + 0] = a0.x;
            As[cur][am * AS_LD + ak + 1] = a0.y;
            As[cur][am * AS_LD + ak + 2] = a0.z;
            As[cur][am * AS_LD + ak + 3] = a0.w;
            As[cur][(am + 64) * AS_LD + ak + 0] = a1.x;
            As[cur][(am + 64) * AS_LD + ak + 1] = a1.y;
            As[cur][(am + 64) * AS_LD + ak + 2] = a1.z;
            As[cur][(am + 64) * AS_LD + ak + 3] = a1.w;
            const int bk = tid >> 4;            // 0..15
            const int bn = (tid & 15) * 4;      // 0..60
            const float4 b4 = *(const float4*)(B + (size_t)(k0 + bk) * ldb + (col0 + bn));
            *(float4*)(&Bs[cur][bk * BS_LD + bn]) = b4;
        }
#endif
        __syncthreads();   // tile (this iteration's buffer) visible to all waves

#pragma unroll
        for (int kk = 0; kk < BK; kk += 4) {
            const int kA = kk + 2 * half;  // lanes 0-15: K=kk,kk+1; lanes 16-31: K=kk+2,kk+3
            v2f a;
            a.x = As[cur][(wave * 16 + ln16) * AS_LD + kA + 0];
            a.y = As[cur][(wave * 16 + ln16) * AS_LD + kA + 1];
#pragma unroll
            for (int nt = 0; nt < 4; nt++) {
                v2f b;
                b.x = Bs[cur][(kA + 0) * BS_LD + nt * 16 + ln16];
                b.y = Bs[cur][(kA + 1) * BS_LD + nt * 16 + ln16];
                acc[nt] = __builtin_amdgcn_wmma_f32_16x16x4_f32(
                    false, a, false, b, (short)0, acc[nt], false, false);
            }
        }
        __syncthreads();   // all reads done before this buffer is refilled
    }

    const float sc = scale_ptr ? scale_ptr[0] : 1.0f;
    const int mrow = row0 + wave * 16;
#pragma unroll
    for (int nt = 0; nt < 4; nt++) {
        const int ncol = col0 + nt * 16 + ln16;
#pragma unroll
        for (int v = 0; v < 8; v++) {
            const int m = mrow + half * 8 + v;   // C layout: vgpr v -> M=v (lanes 0-15), v+8 (16-31)
            float val = acc[nt][v];
            if (bias) val += bias[ncol];
            val *= sc;
            if (residual) val += residual[(size_t)m * ldr + ncol];
            const size_t ci = (size_t)m * ldc + ncol;
            if (flags & 1) val += C[ci];
            if (flags & 2) val = gelu_f(val);
            C[ci] = val;
        }
    }
}

static inline void launch_gemm(hipStream_t st, const float* A, int lda,
                               const float* B, int ldb, float* C, int ldc,
                               const float* bias, const float* res, int ldr,
                               const float* scale, int M, int Ncols, int K, int flags)
{
    dim3 grid(Ncols / BN, M / BM), block(256);
    gemm_wmma_f32<<<grid, block, 0, st>>>(A, lda, B, ldb, C, ldc, bias, res, ldr, scale, K, flags);
}

// ---------------------------------------------------------------------------
// Small kernels
// ---------------------------------------------------------------------------
__global__ void zero_f32(float* p, long n) {
    long i = (long)blockIdx.x * blockDim.x + threadIdx.x;
    if (i < n) p[i] = 0.0f;
}

// W[r,i,o] = sum_b comp[r,b] * bases[b,i,o]
__global__ void rgcn_basis(const float* __restrict__ bases, const float* __restrict__ comp,
                           float* __restrict__ W, int sz)
{
    __shared__ float sc[NREL * NBASES];
    for (int i = threadIdx.x; i < NREL * NBASES; i += blockDim.x) sc[i] = comp[i];
    __syncthreads();
    int j = blockIdx.x * blockDim.x + threadIdx.x;
    if (j >= sz) return;
    float acc[NREL] = {0.f, 0.f, 0.f, 0.f, 0.f};
    for (int b = 0; b < NBASES; b++) {
        float v = bases[(size_t)b * sz + j];
#pragma unroll
        for (int r = 0; r < NREL; r++) acc[r] += sc[r * NBASES + b] * v;
    }
#pragma unroll
    for (int r = 0; r < NREL; r++) W[(size_t)r * sz + j] = acc[r];
}

__global__ void rel_counts(const int* __restrict__ dst, const int* __restrict__ et,
                           float* __restrict__ cnt, int E)
{
    int e = blockIdx.x * blockDim.x + threadIdx.x;
    if (e >= E) return;
    atomicAdd(&cnt[et[e] * NNODES + dst[e]], 1.0f);
}

__global__ void invert_counts(float* c, int n) {
    int i = blockIdx.x * blockDim.x + threadIdx.x;
    if (i < n) c[i] = 1.0f / fmaxf(c[i], 1.0f);
}

// out[dst,c] += h[src,c] * invcnt[dst]   for edges of relation r  (C = 1<<csh)
__global__ void scatter_edges(const float* __restrict__ h, const int* __restrict__ src,
                              const int* __restrict__ dst, const int* __restrict__ et,
                              const float* __restrict__ invc, int r,
                              float* __restrict__ out, int csh, int E)
{
    long idx = (long)blockIdx.x * blockDim.x + threadIdx.x;
    long tot = (long)E << csh;
    if (idx >= tot) return;
    int e = (int)(idx >> csh);
    int c = (int)(idx & ((1 << csh) - 1));
    if (et[e] != r) return;
    int d = dst[e];
    atomicAdd(&out[((size_t)d << csh) + c], h[((size_t)src[e] << csh) + c] * invc[d]);
}

// GraphNorm column statistics (per channel over all nodes)
__global__ void col_stats(const float* __restrict__ X, const float* __restrict__ alpha,
                          float* __restrict__ mu, float* __restrict__ var, int Nn, int C)
{
    int c = blockIdx.x * blockDim.x + threadIdx.x;
    if (c >= C) return;
    float s = 0.f;
    for (int n = 0; n < Nn; n++) s += X[(size_t)n * C + c];
    float m = s / Nn;
    float a = alpha[c], sv = 0.f;
    for (int n = 0; n < Nn; n++) {
        float xc = X[(size_t)n * C + c] - a * m;
        sv += xc * xc;
    }
    mu[c] = m;
    var[c] = sv / Nn;
}

__global__ void gnorm_gelu(const float* __restrict__ X, const float* __restrict__ mu,
                           const float* __restrict__ var, const float* __restrict__ alpha,
                           const float* __restrict__ w, const float* __restrict__ b,
                           float* __restrict__ Y, long total, int csh)
{
    long idx = (long)blockIdx.x * blockDim.x + threadIdx.x;
    if (idx >= total) return;
    int c = (int)(idx & ((1 << csh) - 1));
    float xc = X[idx] - alpha[c] * mu[c];
    float y = xc * rsqrtf(var[c] + EPSV) * w[c] + b[c];
    Y[idx] = gelu_f(y);
}

// Row-wise LayerNorm over C channels (one block per row); bit0 of flags = gelu
__global__ __launch_bounds__(256) void layernorm_act(const float* __restrict__ X,
                                                     const float* __restrict__ g,
                                                     const float* __restrict__ b,
                                                     float* __restrict__ Y, int C, int flags)
{
    const int row = blockIdx.x, tid = threadIdx.x;
    __shared__ float red[256];
    const float* x = X + (size_t)row * C;
    float s = 0.f;
    for (int c = tid; c < C; c += 256) s += x[c];
    red[tid] = s; __syncthreads();
    for (int w = 128; w > 0; w >>= 1) { if (tid < w) red[tid] += red[tid + w]; __syncthreads(); }
    float mean = red[0] / C; __syncthreads();
    float sv = 0.f;
    for (int c = tid; c < C; c += 256) { float d = x[c] - mean; sv += d * d; }
    red[tid] = sv; __syncthreads();
    for (int w = 128; w > 0; w >>= 1) { if (tid < w) red[tid] += red[tid + w]; __syncthreads(); }
    float inv = rsqrtf(red[0] / C + EPSV);
    for (int c = tid; c < C; c += 256) {
        float y = (x[c] - mean) * inv * g[c] + b[c];
        if (flags & 1) y = gelu_f(y);
        Y[(size_t)row * C + c] = y;
    }
}

__global__ void softmax4(const float* __restrict__ sw, float* __restrict__ o) {
    float m = fmaxf(fmaxf(sw[0], sw[1]), fmaxf(sw[2], sw[3]));
    float e0 = expf(sw[0] - m), e1 = expf(sw[1] - m), e2 = expf(sw[2] - m), e3 = expf(sw[3] - m);
    float s = e0 + e1 + e2 + e3;
    o[0] = e0 / s; o[1] = e1 / s; o[2] = e2 / s; o[3] = e3 / s;
}

__global__ void apply_mask(const float* __restrict__ plm, const unsigned char* __restrict__ mask,
                           float* __restrict__ out)
{
    long idx = (long)blockIdx.x * blockDim.x + threadIdx.x;
    if (idx >= (long)NNODES * PHD) return;
    int n = (int)(idx / PHD);
    out[idx] = mask[n] ? plm[idx] : 0.0f;
}

// Cross attention: one block per (query row, head). Exact softmax, scores in LDS.
__global__ __launch_bounds__(128) void cross_attn(const float* __restrict__ Q,
                                                  const float* __restrict__ K,
                                                  const float* __restrict__ V,
                                                  float* __restrict__ O)
{
    const int row = blockIdx.x, h = blockIdx.y, tid = threadIdx.x;
    __shared__ float qs[DH];
    __shared__ float scs[NNODES];
    __shared__ float red[128];

    for (int d = tid; d < DH; d += 128) qs[d] = Q[(size_t)row * PHD + h * DH + d];
    __syncthreads();

    float lmax = -3.4e38f;
    for (int j = tid; j < NNODES; j += 128) {
        const float* kr = K + (size_t)j * PHD + h * DH;
        float s = 0.f;
#pragma unroll 8
        for (int d = 0; d < DH; d++) s += qs[d] * kr[d];
        s *= 0.10206207261596575f;  // 96^-0.5
        scs[j] = s;
        lmax = fmaxf(lmax, s);
    }
    red[tid] = lmax; __syncthreads();
    for (int w = 64; w > 0; w >>= 1) { if (tid < w) red[tid] = fmaxf(red[tid], red[tid + w]); __syncthreads(); }
    float m = red[0]; __syncthreads();

    float lsum = 0.f;
    for (int j = tid; j < NNODES; j += 128) { float e = expf(scs[j] - m); scs[j] = e; lsum += e; }
    red[tid] = lsum; __syncthreads();
    for (int w = 64; w > 0; w >>= 1) { if (tid < w) red[tid] += red[tid + w]; __syncthreads(); }
    float inv = 1.0f / red[0]; __syncthreads();

    for (int d = tid; d < DH; d += 128) {
        float o = 0.f;
        for (int j = 0; j < NNODES; j++) o += scs[j] * V[(size_t)j * PHD + h * DH + d];
        O[(size_t)row * PHD + h * DH + d] = o * inv;
    }
}

__global__ void final_dense(const float* __restrict__ H, const float* __restrict__ W,
                            const float* __restrict__ b, float* __restrict__ out)
{
    int idx = blockIdx.x * blockDim.x + threadIdx.x;
    if (idx >= NNODES * NCLS) return;
    int n = idx / NCLS, c = idx % NCLS;
    float s = b[c];
    const float* h = H + (size_t)n * HCH;
    for (int k = 0; k < HCH; k++) s += h[k] * W[k * NCLS + c];
    out[idx] = s;
}

// ---------------------------------------------------------------------------
// Host orchestration helpers
// ---------------------------------------------------------------------------
static inline void launch_zero(hipStream_t st, float* p, long n) {
    zero_f32<<<dim3((unsigned)((n + 255) / 256)), dim3(256), 0, st>>>(p, n);
}

static void rgcn_block(hipStream_t st, void* const* d_in, int pbase,
                       const float* xin, int di, int dO,
                       float* Wbuf, float* hbuf, float* tmp,
                       float* mu, float* var, const float* invcnt,
                       const int* src, const int* dst, const int* et, float* xout)
{
    const float* bases = (const float*)d_in[pbase + 0];
    const float* comp  = (const float*)d_in[pbase + 1];
    const float* root  = (const float*)d_in[pbase + 2];
    const float* bias  = (const float*)d_in[pbase + 3];
    const float* gw    = (const float*)d_in[pbase + 4];
    const float* gb    = (const float*)d_in[pbase + 5];
    const float* alpha = (const float*)d_in[pbase + 6];

    const int sz = di * dO;
    rgcn_basis<<<dim3((sz + 255) / 256), dim3(256), 0, st>>>(bases, comp, Wbuf, sz);
    // out = x @ root + bias
    launch_gemm(st, xin, di, root, dO, tmp, dO, bias, nullptr, 0, nullptr, NNODES, dO, di, 0);
    const int csh = __builtin_ctz(dO);
    for (int r = 0; r < NREL; r++) {
        launch_gemm(st, xin, di, Wbuf + (size_t)r * sz, dO, hbuf, dO,
                    nullptr, nullptr, 0, nullptr, NNODES, dO, di, 0);
        long tot = (long)NEDGES << csh;
        scatter_edges<<<dim3((unsigned)((tot + 255) / 256)), dim3(256), 0, st>>>(
            hbuf, src, dst, et, invcnt + r * NNODES, r, tmp, csh, NEDGES);
    }
    col_stats<<<dim3((dO + 255) / 256), dim3(256), 0, st>>>(tmp, alpha, mu, var, NNODES, dO);
    long total = (long)NNODES * dO;
    gnorm_gelu<<<dim3((unsigned)((total + 255) / 256)), dim3(256), 0, st>>>(
        tmp, mu, var, alpha, gw, gb, xout, total, csh);
}

extern "C" void kernel_launch(void* const* d_in, const int* in_sizes, int n_in,
                              void* d_out, int out_size, void* d_ws, size_t ws_size,
                              hipStream_t stream)
{
    (void)in_sizes; (void)n_in; (void)out_size; (void)ws_size;
    const float* X   = (const float*)d_in[0];
    const int*   EI  = (const int*)d_in[1];
    const int*   ET  = (const int*)d_in[2];
    const float* PLM = (const float*)d_in[3];
    const unsigned char* MSK = (const unsigned char*)d_in[4];
    const int* SRC = EI;
    const int* DST = EI + NEDGES;
    float* OUT = (float*)d_out;
    auto F = [&](int i) { return (const float*)d_in[i]; };

    // Workspace layout
    char* ws = (char*)d_ws;
    size_t off = 0;
    auto alloc = [&](size_t bytes) -> float* {
        float* p = (float*)(ws + off);
        off = (off + bytes + 255) & ~(size_t)255;
        return p;
    };
    float* smw    = alloc(4 * 4);
    float* invcnt = alloc((size_t)NREL * NNODES * 4);
    float* mu     = alloc((size_t)2048 * 4);
    float* var    = alloc((size_t)2048 * 4);
    float* Wbuf   = alloc((size_t)NREL * 1024 * 2048 * 4);
    float* hbuf   = alloc((size_t)NNODES * 2048 * 4);
    float* tmp    = alloc((size_t)NNODES * 2048 * 4);
    float* x1     = alloc((size_t)NNODES * 256 * 4);
    float* x1r    = alloc((size_t)NNODES * 256 * 4);
    float* x2     = alloc((size_t)NNODES * 512 * 4);
    float* x2r    = alloc((size_t)NNODES * 512 * 4);
    float* x3     = alloc((size_t)NNODES * 1024 * 4);
    float* x4     = alloc((size_t)NNODES * 2048 * 4);
    float* fused  = alloc((size_t)NNODES * PHD * 4);
    float* gnn    = alloc((size_t)NNODES * PHD * 4);
    float* plm_m  = alloc((size_t)NNODES * PHD * 4);
    float* qb     = alloc((size_t)NNODES * PHD * 4);
    float* kb     = alloc((size_t)NNODES * PHD * 4);
    float* vb     = alloc((size_t)NNODES * PHD * 4);
    float* attb   = alloc((size_t)NNODES * PHD * 4);
    float* gout   = alloc((size_t)NNODES * PHD * 4);
    float* tout   = alloc((size_t)NNODES * PHD * 4);
    float* fbuf   = alloc((size_t)NNODES * PHD * 4);
    float* f2     = alloc((size_t)NNODES * PHD * 4);
    float* hclf   = alloc((size_t)NNODES * HCH * 4);

    // Per-relation in-degree reciprocals (shared by all blocks)
    launch_zero(stream, invcnt, (long)NREL * NNODES);
    rel_counts<<<dim3(NEDGES / 256), dim3(256), 0, stream>>>(DST, ET, invcnt, NEDGES);
    invert_counts<<<dim3((NREL * NNODES + 255) / 256), dim3(256), 0, stream>>>(invcnt, NREL * NNODES);

    // --- RGCN blocks ---
    rgcn_block(stream, d_in, RGBASE + 0 * 7, X,   256, 256,  Wbuf, hbuf, tmp, mu, var, invcnt, SRC, DST, ET, x1);
    // x1r = x1 + X @ res1.W + res1.b
    launch_gemm(stream, X, 256, F(RES1W), 256, x1r, 256, F(RES1W + 1), x1, 256, nullptr, NNODES, 256, 256, 0);
    rgcn_block(stream, d_in, RGBASE + 1 * 7, x1r, 256, 512,  Wbuf, hbuf, tmp, mu, var, invcnt, SRC, DST, ET, x2);
    launch_gemm(stream, x1r, 256, F(RES2W), 512, x2r, 512, F(RES2W + 1), x2, 512, nullptr, NNODES, 512, 256, 0);
    rgcn_block(stream, d_in, RGBASE + 2 * 7, x2r, 512, 1024, Wbuf, hbuf, tmp, mu, var, invcnt, SRC, DST, ET, x3);
    rgcn_block(stream, d_in, RGBASE + 3 * 7, x3, 1024, 2048, Wbuf, hbuf, tmp, mu, var, invcnt, SRC, DST, ET, x4);

    // --- Multi-scale fusion: fused = sum_i softmax(sw)[i] * (emb_i @ Wp_i + bp_i) ---
    softmax4<<<dim3(1), dim3(1), 0, stream>>>(F(MSF_SW), smw);
    launch_zero(stream, fused, (long)NNODES * PHD);
    const float* embs[4] = {x1, x2, x3, x4};
    const int    edims[4] = {256, 512, 1024, 2048};
    for (int i = 0; i < 4; i++) {
        launch_gemm(stream, embs[i], edims[i], F(MSF_P0 + 2 * i), PHD, fused, PHD,
                    F(MSF_P0 + 2 * i + 1), nullptr, 0, smw + i, NNODES, PHD, edims[i], 1);
    }
    layernorm_act<<<dim3(NNODES), dim3(256), 0, stream>>>(fused, F(MSF_G), F(MSF_B), gnn, PHD, 0);

    // plm masked
    apply_mask<<<dim3((NNODES * PHD + 255) / 256), dim3(256), 0, stream>>>(PLM, MSK, plm_m);

    // --- Cross attention g2t: queries=gnn, keys/values=plm_m ---
    launch_gemm(stream, gnn,   PHD, F(G2T + 0), PHD, qb, PHD, F(G2T + 1), nullptr, 0, nullptr, NNODES, PHD, PHD, 0);
    launch_gemm(stream, plm_m, PHD, F(G2T + 2), PHD, kb, PHD, F(G2T + 3), nullptr, 0, nullptr, NNODES, PHD, PHD, 0);
    launch_gemm(stream, plm_m, PHD, F(G2T + 4), PHD, vb, PHD, F(G2T + 5), nullptr, 0, nullptr, NNODES, PHD, PHD, 0);
    cross_attn<<<dim3(NNODES, NHEADS), dim3(128), 0, stream>>>(qb, kb, vb, attb);
    launch_gemm(stream, attb, PHD, F(G2T + 6), PHD, gout, PHD, F(G2T + 7), nullptr, 0, nullptr, NNODES, PHD, PHD, 0);

    // --- Cross attention t2g: queries=plm_m, keys/values=gnn ---
    launch_gemm(stream, plm_m, PHD, F(T2G + 0), PHD, qb, PHD, F(T2G + 1), nullptr, 0, nullptr, NNODES, PHD, PHD, 0);
    launch_gemm(stream, gnn,   PHD, F(T2G + 2), PHD, kb, PHD, F(T2G + 3), nullptr, 0, nullptr, NNODES, PHD, PHD, 0);
    launch_gemm(stream, gnn,   PHD, F(T2G + 4), PHD, vb, PHD, F(T2G + 5), nullptr, 0, nullptr, NNODES, PHD, PHD, 0);
    cross_attn<<<dim3(NNODES, NHEADS), dim3(128), 0, stream>>>(qb, kb, vb, attb);
    launch_gemm(stream, attb, PHD, F(T2G + 6), PHD, tout, PHD, F(T2G + 7), nullptr, 0, nullptr, NNODES, PHD, PHD, 0);

    // --- Fusion: f = concat(g,t) @ Wf + bf  (split into two GEMMs) ---
    launch_gemm(stream, gout, PHD, F(FUSW), PHD, fbuf, PHD, F(FUSB), nullptr, 0, nullptr, NNODES, PHD, PHD, 0);
    launch_gemm(stream, tout, PHD, F(FUSW) + (size_t)PHD * PHD, PHD, fbuf, PHD,
                nullptr, nullptr, 0, nullptr, NNODES, PHD, PHD, 1);
    // f = gelu(LN(f))
    layernorm_act<<<dim3(NNODES), dim3(256), 0, stream>>>(fbuf, F(FLNG), F(FLNB), f2, PHD, 1);
    // h = gelu(f @ clf1 + b)
    launch_gemm(stream, f2, PHD, F(C1W), HCH, hclf, HCH, F(C1B), nullptr, 0, nullptr, NNODES, HCH, PHD, 2);
    // out = h @ clf2 + b
    final_dense<<<dim3((NNODES * NCLS + 255) / 256), dim3(256), 0, stream>>>(hclf, F(C2W), F(C2B), OUT);
}